// RKNLayer_45715631899495
// MI455X (gfx1250) — compile-verified
//
#include <hip/hip_runtime.h>
#include <hip/hip_bf16.h>
#include <utility>

// Problem dims (fixed by reference)
#define T_DIM 512
#define B_DIM 32
#define IN_DIM 256
#define H_DIM 256
#define K_DIM 10
#define N_DIM (H_DIM * K_DIM)      // 2560
#define M_ROWS (T_DIM * B_DIM)     // 16384
#define GAP 0.5f
#define ALPHA 0.4f                 // (1/0.5^2)/K
#define EPS 1e-4f
#define TCHUNK 64
#define NCHUNKS (T_DIM / TCHUNK)
#define NS_ITERS 16

typedef float v2f __attribute__((ext_vector_type(2)));
typedef float v8f __attribute__((ext_vector_type(8)));

// ---------------- WMMA f32 16x16x4 register-blocked tile GEMM ----------------
// A operand: lane l(<16) holds row m0+l, K = kk + 2*(lane>>4) + {0,1}
// B operand (NT): lane l holds row n0+l of B-rows, same K offsets
// B operand (NN): lane l holds col n0+l, rows kk+koff, kk+koff+1
__device__ __forceinline__ v8f wmma_acc(v8f c, v2f a, v2f b) {
    return __builtin_amdgcn_wmma_f32_16x16x4_f32(
        /*neg_a=*/false, a, /*neg_b=*/false, b,
        /*c_mod=*/(short)0, c, /*reuse_a=*/false, /*reuse_b=*/false);
}

template <bool TRANS_B, int MB, int NB>
__device__ __forceinline__ void tile_gemm_blk(const float* __restrict__ A, int lda,
                                              const float* __restrict__ B, int ldb,
                                              int Kd, int m0, int n0,
                                              v8f (&c)[MB][NB]) {
    const int lane = threadIdx.x & 31;
    const int l    = lane & 15;
    const int koff = (lane >> 4) << 1;
    const float* arow[MB];
#pragma unroll
    for (int mi = 0; mi < MB; ++mi)
        arow[mi] = A + (size_t)(m0 + mi * 16 + l) * lda;
    const float* brow[NB];
#pragma unroll
    for (int nj = 0; nj < NB; ++nj)
        brow[nj] = TRANS_B ? (B + (size_t)(n0 + nj * 16 + l) * ldb) : B;

    for (int kk = 0; kk < Kd; kk += 4) {
        const int ka = kk + koff;
        v2f a[MB], b[NB];
#pragma unroll
        for (int mi = 0; mi < MB; ++mi)
            a[mi] = *(const v2f*)(arow[mi] + ka);
#pragma unroll
        for (int nj = 0; nj < NB; ++nj) {
            if (TRANS_B) {
                b[nj] = *(const v2f*)(brow[nj] + ka);
            } else {
                b[nj].x = B[(size_t)ka * ldb + n0 + nj * 16 + l];
                b[nj].y = B[(size_t)(ka + 1) * ldb + n0 + nj * 16 + l];
            }
        }
#pragma unroll
        for (int mi = 0; mi < MB; ++mi)
#pragma unroll
            for (int nj = 0; nj < NB; ++nj)
                c[mi][nj] = wmma_acc(c[mi][nj], a[mi], b[nj]);
    }
}

template <int MB, int NB>
__device__ __forceinline__ void zero_acc(v8f (&c)[MB][NB]) {
#pragma unroll
    for (int mi = 0; mi < MB; ++mi)
#pragma unroll
        for (int nj = 0; nj < NB; ++nj)
            c[mi][nj] = (v8f){0.f, 0.f, 0.f, 0.f, 0.f, 0.f, 0.f, 0.f};
}

// ---------------- elementwise / reduction kernels ----------------
__global__ void k_wnorm(const float* __restrict__ w, float* __restrict__ wn) {
    __shared__ float red[256];
    const int row = blockIdx.x;            // 0..N_DIM-1
    const int tid = threadIdx.x;           // 256 = IN_DIM
    float v = w[(size_t)row * IN_DIM + tid];
    red[tid] = v * v;
    __syncthreads();
    for (int s = 128; s > 0; s >>= 1) {
        if (tid < s) red[tid] += red[tid + s];
        __syncthreads();
    }
    float d = fmaxf(sqrtf(red[0]), EPS);
    wn[(size_t)row * IN_DIM + tid] = v / d;
}

__global__ void k_xnorm(const float* __restrict__ x, float* __restrict__ xn) {
    __shared__ float red[256];
    const int row = blockIdx.x;            // 0..M_ROWS-1
    const int tid = threadIdx.x;
    float v = x[(size_t)row * IN_DIM + tid];
    red[tid] = v * v;
    __syncthreads();
    for (int s = 128; s > 0; s >>= 1) {
        if (tid < s) red[tid] += red[tid + s];
        __syncthreads();
    }
    if (tid == 0) xn[row] = sqrtf(red[0]);
}

// gram-product matrix: M[h][g] = exp(ALPHA * (dot_{2560}(wn_h, wn_g) - K))
// 32x32 tile per wave (MB=2, NB=2), NT, Kd = 2560
__global__ void k_gram(const float* __restrict__ wn, float* __restrict__ Mg) {
    const int wid = blockIdx.x * (blockDim.x >> 5) + (threadIdx.x >> 5);
    const int tilesN = H_DIM / 32;               // 8
    if (wid >= (H_DIM / 32) * tilesN) return;    // wave-uniform
    const int m0 = (wid / tilesN) * 32, n0 = (wid % tilesN) * 32;
    v8f c[2][2];
    zero_acc(c);
    tile_gemm_blk<true, 2, 2>(wn, N_DIM, wn, N_DIM, N_DIM, m0, n0, c);
    const int lane = threadIdx.x & 31, l = lane & 15, rb = (lane >> 4) * 8;
#pragma unroll
    for (int mi = 0; mi < 2; ++mi)
#pragma unroll
        for (int nj = 0; nj < 2; ++nj)
#pragma unroll
            for (int v = 0; v < 8; v++)
                Mg[(size_t)(m0 + mi * 16 + v + rb) * H_DIM + n0 + nj * 16 + l] =
                    expf(ALPHA * (c[mi][nj][v] - (float)K_DIM));
}

// b[t,b,h,k] chunk: NT GEMM + norm*exp(alpha*(dot/max(norm,eps)-1)) epilogue
// 32x64 tile per wave (MB=2, NB=4)
__global__ void k_bgemm(const float* __restrict__ x, const float* __restrict__ wn,
                        const float* __restrict__ xnorm, float* __restrict__ bout,
                        int rowBase) {
    const int wid = blockIdx.x * (blockDim.x >> 5) + (threadIdx.x >> 5);
    const int tilesN = N_DIM / 64;                  // 40
    const int tilesM = (TCHUNK * B_DIM) / 32;       // 64
    if (wid >= tilesM * tilesN) return;             // wave-uniform
    const int m0 = (wid / tilesN) * 32, n0 = (wid % tilesN) * 64;
    v8f c[2][4];
    zero_acc(c);
    tile_gemm_blk<true, 2, 4>(x + (size_t)rowBase * IN_DIM, IN_DIM,
                              wn, IN_DIM, IN_DIM, m0, n0, c);
    const int lane = threadIdx.x & 31, l = lane & 15, rb = (lane >> 4) * 8;
#pragma unroll
    for (int mi = 0; mi < 2; ++mi)
#pragma unroll
        for (int v = 0; v < 8; v++) {
            const int r = m0 + mi * 16 + v + rb;    // local chunk row
            const float nrm = xnorm[rowBase + r];
            const float d = fmaxf(nrm, EPS);
#pragma unroll
            for (int nj = 0; nj < 4; ++nj)
                bout[(size_t)r * N_DIM + n0 + nj * 16 + l] =
                    nrm * expf(ALPHA * (c[mi][nj][v] / d - 1.0f));
        }
}

// generic NN GEMM (32x32 tiles) with optional alpha = rsqrt(sqrt(*sptr)) * cmul
__global__ void k_gemm_nn(const float* __restrict__ A, const float* __restrict__ B,
                          float* __restrict__ C, int Mr, int Nc, int Kd,
                          const float* __restrict__ sptr, float cmul) {
    const int wid = blockIdx.x * (blockDim.x >> 5) + (threadIdx.x >> 5);
    const int tilesN = Nc / 32;
    if (wid >= (Mr / 32) * tilesN) return;          // wave-uniform
    const int m0 = (wid / tilesN) * 32, n0 = (wid % tilesN) * 32;
    v8f c[2][2];
    zero_acc(c);
    tile_gemm_blk<false, 2, 2>(A, Kd, B, Nc, Kd, m0, n0, c);
    float alpha = cmul;
    if (sptr) alpha *= rsqrtf(sqrtf(*sptr));        // M^{-1/2} = Z / sqrt(||M||_F)
    const int lane = threadIdx.x & 31, l = lane & 15, rb = (lane >> 4) * 8;
#pragma unroll
    for (int mi = 0; mi < 2; ++mi)
#pragma unroll
        for (int nj = 0; nj < 2; ++nj)
#pragma unroll
            for (int v = 0; v < 8; v++)
                C[(size_t)(m0 + mi * 16 + v + rb) * Nc + n0 + nj * 16 + l] =
                    alpha * c[mi][nj][v];
}

__global__ void k_sumsq(const float* __restrict__ Mg, float* __restrict__ sumsq) {
    __shared__ float red[256];
    const int idx = blockIdx.x * 256 + threadIdx.x;
    float v = Mg[idx];
    red[threadIdx.x] = v * v;
    __syncthreads();
    for (int s = 128; s > 0; s >>= 1) {
        if (threadIdx.x < s) red[threadIdx.x] += red[threadIdx.x + s];
        __syncthreads();
    }
    if (threadIdx.x == 0) atomicAdd(sumsq, red[0]);
}

__global__ void k_nsinit(const float* __restrict__ Mg, const float* __restrict__ sumsq,
                         float* __restrict__ Y, float* __restrict__ Z) {
    const int idx = blockIdx.x * 256 + threadIdx.x;   // H*H
    const float cfrob = sqrtf(*sumsq);
    const int r = idx / H_DIM, cc = idx % H_DIM;
    Y[idx] = Mg[idx] / cfrob;
    Z[idx] = (r == cc) ? 1.f : 0.f;
}

// P <- 1.5*I - 0.5*P   (Newton-Schulz T-matrix, in place)
__global__ void k_nstm(float* __restrict__ P) {
    const int idx = blockIdx.x * 256 + threadIdx.x;
    const int r = idx / H_DIM, cc = idx % H_DIM;
    P[idx] = ((r == cc) ? 1.5f : 0.f) - 0.5f * P[idx];
}

// recurrence chunk: c_t[j] = GAP*c[j] + c[j-1]*b_t[j], c[-1]:=1
__global__ void k_recur(const float* __restrict__ bchunk, float* __restrict__ state,
                        float* __restrict__ meanacc, float* __restrict__ cs_last,
                        int tBase) {
    const int tid = blockIdx.x * blockDim.x + threadIdx.x;  // 0..B*H-1
    if (tid >= B_DIM * H_DIM) return;
    const int bb = tid / H_DIM, h = tid % H_DIM;
    float c[K_DIM];
    float* st = state + (size_t)tid * K_DIM;                // (B,H,K)
#pragma unroll
    for (int k = 0; k < K_DIM; k++) c[k] = st[k];
    float macc = 0.f;
    for (int t = 0; t < TCHUNK; ++t) {
        const float* bp = bchunk + ((size_t)t * B_DIM + bb) * N_DIM + h * K_DIM;
        float bv[K_DIM];
#pragma unroll
        for (int k = 0; k < K_DIM; k++) bv[k] = bp[k];
#pragma unroll
        for (int j = K_DIM - 1; j >= 1; --j) c[j] = GAP * c[j] + c[j - 1] * bv[j];
        c[0] = GAP * c[0] + bv[0];
        const float last = c[K_DIM - 1];
        cs_last[((size_t)(tBase + t) * B_DIM + bb) * H_DIM + h] = last;
        macc += last;
    }
#pragma unroll
    for (int k = 0; k < K_DIM; k++) st[k] = c[k];
    meanacc[tid] += macc;
}

__global__ void k_copy(const float* __restrict__ src, float* __restrict__ dst, int n) {
    const int i = blockIdx.x * blockDim.x + threadIdx.x;
    if (i < n) dst[i] = src[i];
}

// ---------------- host orchestration ----------------
extern "C" void kernel_launch(void* const* d_in, const int* in_sizes, int n_in,
                              void* d_out, int out_size, void* d_ws, size_t ws_size,
                              hipStream_t stream) {
    (void)in_sizes; (void)n_in; (void)out_size; (void)ws_size;
    const float* x = (const float*)d_in[0];   // (T,B,IN)
    const float* w = (const float*)d_in[1];   // (H,K,IN)
    float* out = (float*)d_out;
    float* out_output  = out;                                     // (B,H)
    float* out_outputs = out + (size_t)B_DIM * H_DIM;             // (T,B,H)
    float* out_hx      = out_outputs + (size_t)T_DIM * B_DIM * H_DIM; // (B,H,K)

    char* ws = (char*)d_ws;
    size_t off = 0;
    auto alloc = [&](size_t nfloats) {
        float* p = (float*)(ws + off);
        off += ((nfloats * sizeof(float)) + 255) & ~(size_t)255;  // keep 256B aligned
        return p;
    };
    float* wn      = alloc((size_t)N_DIM * IN_DIM);       // 2.6 MB
    float* xnorm   = alloc(M_ROWS);
    float* Mg      = alloc((size_t)H_DIM * H_DIM);
    float* Y       = alloc((size_t)H_DIM * H_DIM);
    float* Z       = alloc((size_t)H_DIM * H_DIM);
    float* Y2      = alloc((size_t)H_DIM * H_DIM);
    float* Z2      = alloc((size_t)H_DIM * H_DIM);
    float* P       = alloc((size_t)H_DIM * H_DIM);
    float* sumsq   = alloc(64);                            // scalar (padded)
    float* state   = alloc((size_t)B_DIM * H_DIM * K_DIM);
    float* meanacc = alloc((size_t)B_DIM * H_DIM);
    float* cs_last = alloc((size_t)T_DIM * B_DIM * H_DIM); // 16.8 MB
    float* bchunk  = alloc((size_t)TCHUNK * B_DIM * N_DIM);// 21 MB

    // 1) normalize weight rows; per-row x norms
    k_wnorm<<<N_DIM, 256, 0, stream>>>(w, wn);
    k_xnorm<<<M_ROWS, 256, 0, stream>>>(x, xnorm);

    // 2) gram product as one 256x256x2560 NT WMMA GEMM + exp epilogue
    {
        int waves = (H_DIM / 32) * (H_DIM / 32);     // 64
        k_gram<<<(waves + 7) / 8, 256, 0, stream>>>(wn, Mg);
    }

    // 3) Newton-Schulz inverse sqrt: Y<-M/c, Z<-I; iterate with 256^3 WMMA GEMMs
    hipMemsetAsync(sumsq, 0, sizeof(float), stream);
    k_sumsq<<<(H_DIM * H_DIM) / 256, 256, 0, stream>>>(Mg, sumsq);
    k_nsinit<<<(H_DIM * H_DIM) / 256, 256, 0, stream>>>(Mg, sumsq, Y, Z);
    for (int it = 0; it < NS_ITERS; ++it) {
        int waves = (H_DIM / 32) * (H_DIM / 32);     // 64
        int blocks = (waves + 7) / 8;
        k_gemm_nn<<<blocks, 256, 0, stream>>>(Z, Y, P, H_DIM, H_DIM, H_DIM, nullptr, 1.f);
        k_nstm<<<(H_DIM * H_DIM) / 256, 256, 0, stream>>>(P);
        k_gemm_nn<<<blocks, 256, 0, stream>>>(Y, P, Y2, H_DIM, H_DIM, H_DIM, nullptr, 1.f);
        k_gemm_nn<<<blocks, 256, 0, stream>>>(P, Z, Z2, H_DIM, H_DIM, H_DIM, nullptr, 1.f);
        std::swap(Y, Y2);
        std::swap(Z, Z2);
    }

    // 4) chunked big GEMM (WMMA f32) + gap recurrence over T
    hipMemsetAsync(state, 0, (size_t)B_DIM * H_DIM * K_DIM * sizeof(float), stream);
    hipMemsetAsync(meanacc, 0, (size_t)B_DIM * H_DIM * sizeof(float), stream);
    for (int ch = 0; ch < NCHUNKS; ++ch) {
        const int rowBase = ch * TCHUNK * B_DIM;
        int waves = ((TCHUNK * B_DIM) / 32) * (N_DIM / 64);   // 64*40 = 2560
        k_bgemm<<<(waves + 7) / 8, 256, 0, stream>>>(x, wn, xnorm, bchunk, rowBase);
        k_recur<<<(B_DIM * H_DIM) / 256, 256, 0, stream>>>(bchunk, state, meanacc,
                                                           cs_last, ch * TCHUNK);
    }

    // 5) whitening: outputs = cs_last @ Z * c^-1/2 ; output = (meanacc/T) @ Z * c^-1/2
    {
        int waves = (M_ROWS / 32) * (H_DIM / 32);    // 512*8 = 4096
        k_gemm_nn<<<(waves + 7) / 8, 256, 0, stream>>>(cs_last, Z, out_outputs,
                                                       M_ROWS, H_DIM, H_DIM, sumsq, 1.0f);
    }
    {
        int waves = (B_DIM / 32) * (H_DIM / 32);     // 8
        k_gemm_nn<<<1, 256, 0, stream>>>(meanacc, Z, out_output,
                                         B_DIM, H_DIM, H_DIM, sumsq,
                                         1.0f / (float)T_DIM);
    }
    // 6) hx_T
    k_copy<<<(B_DIM * H_DIM * K_DIM + 255) / 256, 256, 0, stream>>>(
        state, out_hx, B_DIM * H_DIM * K_DIM);
}